// Gemma4MoE_70248485093993
// MI455X (gfx1250) — compile-verified
//
#include <hip/hip_runtime.h>
#include <math.h>

typedef __attribute__((ext_vector_type(2))) float v2f;
typedef __attribute__((ext_vector_type(8))) float v8f;

#define TT 2048   // tokens
#define HH 2048   // hidden
#define EE 8      // experts
#define II 2048   // intermediate

#define MT 32           // tokens per block (2 WMMA M-tiles)
#define NBW 64          // n-block width handled per wave iteration (4 WMMA N-tiles)
#define NB_PER_WAVE 4   // 4 * 64 = 256 columns per wave; 8 waves cover 2048

static __device__ __forceinline__ v8f wmma_f32(v2f a, v2f b, v8f c) {
    // V_WMMA_F32_16X16X4_F32 : D = A(16x4) x B(4x16) + C(16x16), fp32
    return __builtin_amdgcn_wmma_f32_16x16x4_f32(
        /*neg_a=*/false, a, /*neg_b=*/false, b,
        /*c_mod=*/(short)0, c, /*reuse_a=*/false, /*reuse_b=*/false);
}

static __device__ __forceinline__ float gelu_tanh(float x) {
    // jax.nn.gelu(approximate=True): 0.5*x*(1+tanh(sqrt(2/pi)*(x+0.044715x^3)))
    //                              = x * sigmoid(2*sqrt(2/pi)*(x+0.044715x^3))
    // one v_exp_f32 + one v_rcp_f32, branch-free (stable at both tails)
    const float c2 = 1.5957691216057308f;  // 2*sqrt(2/pi)
    float z = -c2 * (x + 0.044715f * x * x * x);
    return x * __builtin_amdgcn_rcpf(1.0f + __expf(z));
}

// ---------------------------------------------------------------------------
// Routing: dense combine[T, E] (zero outside top-2 slots)
// ---------------------------------------------------------------------------
__global__ void route_kernel(const float* __restrict__ logits,
                             const float* __restrict__ scale,
                             float* __restrict__ combine) {
    int t = blockIdx.x * blockDim.x + threadIdx.x;
    if (t >= TT) return;
    float l[EE];
    float mx = -INFINITY;
    #pragma unroll
    for (int e = 0; e < EE; ++e) { l[e] = logits[t * EE + e]; mx = fmaxf(mx, l[e]); }
    float p[EE]; float denom = 0.f;
    #pragma unroll
    for (int e = 0; e < EE; ++e) { p[e] = expf(l[e] - mx); denom += p[e]; }
    // top-2 over logits, first-index tie-break (matches lax.top_k)
    int i1 = 0;
    #pragma unroll
    for (int e = 1; e < EE; ++e) if (l[e] > l[i1]) i1 = e;
    int i2 = (i1 == 0) ? 1 : 0;
    #pragma unroll
    for (int e = 0; e < EE; ++e) if (e != i1 && l[e] > l[i2]) i2 = e;
    float p1 = p[i1] / denom, p2 = p[i2] / denom;
    float renorm = p1 + p2;
    if (!(renorm > 0.f)) renorm = 1.f;
    float w1 = (p1 / renorm) * scale[i1];
    float w2 = (p2 / renorm) * scale[i2];
    #pragma unroll
    for (int e = 0; e < EE; ++e) combine[t * EE + e] = 0.f;
    combine[t * EE + i1] = w1;
    combine[t * EE + i2] = w2;
}

__global__ void zero_out_kernel(float* __restrict__ out, int n) {
    int i = blockIdx.x * blockDim.x + threadIdx.x;
    if (i < n) out[i] = 0.f;
}

// ---------------------------------------------------------------------------
// Fused expert MLP: block = (expert e, 32-token tile). Phase 1 computes
// act[32, I] = combine * gelu(x Wg) * (x Wu) into LDS (256 KB, CDNA5 only);
// Phase 2 computes act @ Wd and atomically accumulates into out.
// ---------------------------------------------------------------------------
__global__ __launch_bounds__(256, 1)
void moe_kernel(const float* __restrict__ x,
                const float* __restrict__ wg,
                const float* __restrict__ wu,
                const float* __restrict__ wd,
                const float* __restrict__ combine,
                float* __restrict__ out) {
    __shared__ float act[MT * II];   // 32 x 2048 fp32 = 256 KB (fits 320 KB WGP LDS)
    __shared__ float cmb[MT];
    __shared__ int any_flag;

    const int e  = blockIdx.x;
    const int t0 = blockIdx.y * MT;

    const int tid  = threadIdx.x;
    const int lane = tid & 31;
    const int wave = tid >> 5;
    const int half = lane >> 4;      // lane-half selects K {0,1} vs {2,3} (A/B frags)
    const int lrow = lane & 15;      // A: M row; B/C: N column
    const int koff = half << 1;

    if (tid == 0) any_flag = 0;
    __syncthreads();
    if (tid < MT) {
        float c = combine[(size_t)(t0 + tid) * EE + e];
        cmb[tid] = c;
        if (c != 0.f) atomicOr(&any_flag, 1);
    }
    __syncthreads();
    if (!any_flag) return;   // uniform: no token in tile routed to this expert

    const float* __restrict__ wge = wg + (size_t)e * HH * II;
    const float* __restrict__ wue = wu + (size_t)e * HH * II;
    const float* __restrict__ wde = wd + (size_t)e * II * HH;

    // ---------------- Phase 1: gate/up GEMMs + gated GELU -> LDS ----------
    for (int nb = 0; nb < NB_PER_WAVE; ++nb) {
        const int n0 = wave * (NB_PER_WAVE * NBW) + nb * NBW;  // I-dim base
        v8f g[2][4] = {};
        v8f u[2][4] = {};
        #pragma unroll 2
        for (int k = 0; k < HH; k += 4) {
            v2f a0 = *(const v2f*)(x + (size_t)(t0 + lrow) * HH + k + koff);
            v2f a1 = *(const v2f*)(x + (size_t)(t0 + 16 + lrow) * HH + k + koff);
            #pragma unroll
            for (int nt = 0; nt < 4; ++nt) {
                const size_t ci = (size_t)(k + koff) * II + (n0 + nt * 16 + lrow);
                v2f bg; bg.x = wge[ci]; bg.y = wge[ci + II];
                v2f bu; bu.x = wue[ci]; bu.y = wue[ci + II];
                g[0][nt] = wmma_f32(a0, bg, g[0][nt]);
                g[1][nt] = wmma_f32(a1, bg, g[1][nt]);
                u[0][nt] = wmma_f32(a0, bu, u[0][nt]);
                u[1][nt] = wmma_f32(a1, bu, u[1][nt]);
            }
        }
        // epilogue: C/D layout -> (m, n); m = mt*16 + r + 8*half, n = lane&15
        #pragma unroll
        for (int mt = 0; mt < 2; ++mt)
            #pragma unroll
            for (int nt = 0; nt < 4; ++nt)
                #pragma unroll
                for (int r = 0; r < 8; ++r) {
                    int m = mt * 16 + r + 8 * half;
                    int n = n0 + nt * 16 + lrow;
                    float gv = gelu_tanh(g[mt][nt][r]);
                    act[m * II + n] = gv * u[mt][nt][r] * cmb[m];
                }
    }
    __syncthreads();

    // ---------------- Phase 2: down GEMM from LDS, atomic accumulate ------
    for (int nb = 0; nb < NB_PER_WAVE; ++nb) {
        const int n0 = wave * (NB_PER_WAVE * NBW) + nb * NBW;  // H-dim base
        v8f o[2][4] = {};
        #pragma unroll 2
        for (int k = 0; k < II; k += 4) {
            v2f a0 = *(const v2f*)(&act[lrow * II + k + koff]);
            v2f a1 = *(const v2f*)(&act[(16 + lrow) * II + k + koff]);
            #pragma unroll
            for (int nt = 0; nt < 4; ++nt) {
                const size_t ci = (size_t)(k + koff) * HH + (n0 + nt * 16 + lrow);
                v2f b; b.x = wde[ci]; b.y = wde[ci + HH];
                o[0][nt] = wmma_f32(a0, b, o[0][nt]);
                o[1][nt] = wmma_f32(a1, b, o[1][nt]);
            }
        }
        #pragma unroll
        for (int mt = 0; mt < 2; ++mt)
            #pragma unroll
            for (int nt = 0; nt < 4; ++nt)
                #pragma unroll
                for (int r = 0; r < 8; ++r) {
                    int m = mt * 16 + r + 8 * half;
                    int n = n0 + nt * 16 + lrow;
                    atomicAdd(&out[(size_t)(t0 + m) * HH + n], o[mt][nt][r]);
                }
    }
}

// ---------------------------------------------------------------------------
extern "C" void kernel_launch(void* const* d_in, const int* in_sizes, int n_in,
                              void* d_out, int out_size, void* d_ws, size_t ws_size,
                              hipStream_t stream) {
    const float* x      = (const float*)d_in[0];  // [T, H]
    const float* logits = (const float*)d_in[1];  // [T, E]
    const float* scale  = (const float*)d_in[2];  // [E]
    const float* wg     = (const float*)d_in[3];  // [E, H, I]
    const float* wu     = (const float*)d_in[4];  // [E, H, I]
    const float* wd     = (const float*)d_in[5];  // [E, I, H]
    float* out          = (float*)d_out;          // [T, H]

    float* combine = (float*)d_ws;                // [T, E] = 64 KB scratch

    route_kernel<<<(TT + 255) / 256, 256, 0, stream>>>(logits, scale, combine);

    const int nout = TT * HH;
    zero_out_kernel<<<(nout + 255) / 256, 256, 0, stream>>>(out, nout);

    dim3 grid(EE, TT / MT);   // expert-major: co-resident blocks share weight stream in L2
    moe_kernel<<<grid, 256, 0, stream>>>(x, wg, wu, wd, combine, out);
}